// CNFPCFG_90056874262850
// MI455X (gfx1250) — compile-verified
//
#include <hip/hip_runtime.h>

#define R      96
#define BATCH  96
#define SEQ    24
#define RR     (R * R)          // 9216

typedef float v2f __attribute__((ext_vector_type(2)));
typedef float v8f __attribute__((ext_vector_type(8)));

// ---------------------------------------------------------------------------
// Kernel 1: rules = softmax(binary_logits over last R*R), stored transposed in
// PAIRED-ROW layout:  rulesP[(yz/2)*2*R + x*2 + (yz&1)] = rules[x][yz]
// so B-fragment rows (k, k+1) for one column x are one aligned b64 load.
// ---------------------------------------------------------------------------
__global__ void cnf_softmax_rules(const float* __restrict__ logits,
                                  float* __restrict__ rulesP) {
    const int x   = blockIdx.x;              // nonterminal 0..95
    const int tid = threadIdx.x;             // 256 threads
    const float* row = logits + (size_t)x * RR;
    __shared__ float red[256];

    float mx = -1e30f;
    for (int k = tid; k < RR; k += 256) mx = fmaxf(mx, row[k]);
    red[tid] = mx; __syncthreads();
    for (int s = 128; s > 0; s >>= 1) {
        if (tid < s) red[tid] = fmaxf(red[tid], red[tid + s]);
        __syncthreads();
    }
    mx = red[0]; __syncthreads();

    float sum = 0.f;
    for (int k = tid; k < RR; k += 256) sum += __expf(row[k] - mx);
    red[tid] = sum; __syncthreads();
    for (int s = 128; s > 0; s >>= 1) {
        if (tid < s) red[tid] += red[tid + s];
        __syncthreads();
    }
    const float inv = 1.0f / red[0];

    for (int k = tid; k < RR; k += 256)
        rulesP[(size_t)(k & ~1) * R + x * 2 + (k & 1)] = __expf(row[k] - mx) * inv;
}

// ---------------------------------------------------------------------------
// Kernel 2: chart diagonal init: chart[b,i,i,x] = (x == token[b][i])
// ---------------------------------------------------------------------------
__global__ void cnf_init_chart(const int* __restrict__ tokens,
                               float* __restrict__ chart) {
    const int cell = blockIdx.x;             // b*SEQ + i
    const int b = cell / SEQ, i = cell % SEQ;
    const int tok = tokens[cell];
    const int x = threadIdx.x;               // 96 threads
    chart[(((size_t)b * SEQ + i) * SEQ + i) * R + x] = (x == tok) ? 1.0f : 0.0f;
}

// ---------------------------------------------------------------------------
// Kernel 3 (per span, SPAN compile-time): one workgroup = 16 cells (M-tile),
// 192 threads = 6 waves, each wave owns a 16-wide N-slice of val[16 x 96].
// ---------------------------------------------------------------------------
template <int SPAN>
__global__ __launch_bounds__(192, 1)
void cnf_span_kernel(const float* __restrict__ rulesP,
                     float* __restrict__ chart) {
    constexpr int P = SEQ - SPAN + 1;        // start positions
    constexpr int K = SPAN - 1;              // split points
    extern __shared__ float smem[];
    const int c0   = blockIdx.x * 16;        // first cell of tile
    const int tid  = threadIdx.x;
    const int lane = tid & 31;
    const int wave = tid >> 5;               // 0..5 -> N-tile

    float* rightL = smem;                    // 16 * K * 96
    float* leftYB = rightL + 16 * K * R;     // 16 * K * 2
    float* pairA  = leftYB + 16 * K * 2;     // 16 * 194 (padded stride)

    const int zz  = tid % R;                 // 0..95  (192 = 2*96)
    const int hh  = tid / R;                 // 0 or 1

    // ---- stage right: right[p][k][z] = chart[b, i+k+1, i+span-1, z] ----
    // idx = tid + 192*t  ->  z = zz (const), row = hh + 2t
    for (int row = hh; row < 16 * K; row += 2) {
        const int p = row / K;
        const int k = row - p * K;
        const int c = c0 + p;
        const int b = c / P, i = c - b * P;
        const int j = i + SPAN - 1;
        rightL[row * R + zz] =
            chart[(((size_t)b * SEQ + (i + k + 1)) * SEQ + j) * R + zz];
    }

    v8f acc = {};                            // 16x16 f32 accumulator
    const int xcol = (wave << 4) + (lane & 15);
    const int koff = (lane < 16) ? 0 : 2;    // A/B K sub-offset per half-wave
    const int prow = lane & 15;              // A-matrix M row
    __syncthreads();

    for (int yb = 0; yb < R / 2; ++yb) {
        const int ybase = yb * 2;
        // ---- stage left slab: left[p][k][y2] = chart[b, i, i+k, ybase+y2] ----
        for (int idx = tid; idx < 16 * K * 2; idx += 192) {
            const int p   = idx / (K * 2);
            const int rem = idx - p * (K * 2);
            const int k   = rem >> 1;
            const int y2  = rem & 1;
            const int c = c0 + p;
            const int b = c / P, i = c - b * P;
            leftYB[idx] =
                chart[(((size_t)b * SEQ + i) * SEQ + (i + k)) * R + (ybase + y2)];
        }
        __syncthreads();

        // ---- build pair block: pairA[p][y2*96+z] = sum_k left*right ----
        // 3072 entries / 192 threads: entry e = it*192 + tid -> p = it, rem = tid
        for (int it = 0; it < 16; ++it) {
            const float* lp = leftYB + it * K * 2 + hh;
            const float* rp = rightL + it * K * R + zz;
            float s = 0.f;
#pragma unroll
            for (int k = 0; k < K; ++k) s += lp[2 * k] * rp[k * R];
            pairA[it * 194 + tid] = s;
        }
        __syncthreads();

        // ---- WMMA: consume K-chunk of 192 (48 steps of K=4) ----
        const int yz0 = ybase * R;           // global yz base (even)
#pragma unroll 8
        for (int kk = 0; kk < 192; kk += 4) {
            const v2f a = *(const v2f*)(pairA + prow * 194 + kk + koff);
            // paired-row layout: rows (r, r+1) adjacent per column -> one b64
            const v2f bf = *(const v2f*)(rulesP +
                (size_t)(yz0 + kk + koff) * R + xcol * 2);
            acc = __builtin_amdgcn_wmma_f32_16x16x4_f32(
                false, a, false, bf, (short)0, acc, false, false);
        }
        __syncthreads();
    }

    // ---- write D: VGPR r -> M = r (lanes 0-15) / r+8 (lanes 16-31) ----
    const int padp = (lane >= 16) ? 8 : 0;
#pragma unroll
    for (int r = 0; r < 8; ++r) {
        const int p = r + padp;
        const int c = c0 + p;
        const int b = c / P, i = c - b * P;
        const int j = i + SPAN - 1;
        chart[(((size_t)b * SEQ + i) * SEQ + j) * R + xcol] = acc[r];
    }
}

// ---------------------------------------------------------------------------
// Kernel 4: out[b] = dot(softmax(start_logits), chart[b, 0, SEQ-1, :])
// ---------------------------------------------------------------------------
__global__ void cnf_finalize(const float* __restrict__ start_logits,
                             const float* __restrict__ chart,
                             float* __restrict__ out) {
    __shared__ float s[R];
    __shared__ float red;
    const int t = threadIdx.x;               // 96 threads, t = batch index
    const float l = start_logits[t];
    s[t] = l; __syncthreads();
    if (t == 0) {
        float mx = -1e30f;
        for (int i = 0; i < R; ++i) mx = fmaxf(mx, s[i]);
        red = mx;
    }
    __syncthreads();
    const float e = __expf(l - red);
    s[t] = e; __syncthreads();
    if (t == 0) {
        float sum = 0.f;
        for (int i = 0; i < R; ++i) sum += s[i];
        red = sum;
    }
    __syncthreads();
    s[t] = e / red; __syncthreads();

    float acc = 0.f;
    const float* cp = chart + (((size_t)t * SEQ + 0) * SEQ + (SEQ - 1)) * R;
    for (int x = 0; x < R; ++x) acc += s[x] * cp[x];
    out[t] = acc;
}

// ---------------------------------------------------------------------------
extern "C" void kernel_launch(void* const* d_in, const int* in_sizes, int n_in,
                              void* d_out, int out_size, void* d_ws, size_t ws_size,
                              hipStream_t stream) {
    (void)in_sizes; (void)n_in; (void)out_size; (void)ws_size;
    const float* binary_logits = (const float*)d_in[0];
    const float* start_logits  = (const float*)d_in[1];
    const int*   tokens        = (const int*)d_in[2];
    float* out = (float*)d_out;

    float* rulesP = (float*)d_ws;                        // RR * R floats (3.5 MB)
    float* chart  = rulesP + (size_t)RR * R;             // BATCH*SEQ*SEQ*R (21 MB)

    cnf_softmax_rules<<<R, 256, 0, stream>>>(binary_logits, rulesP);
    cnf_init_chart<<<BATCH * SEQ, R, 0, stream>>>(tokens, chart);

#define LAUNCH_SPAN(S)                                                        \
    case S: {                                                                 \
        constexpr int K_ = S - 1;                                             \
        const size_t shmem =                                                  \
            (size_t)(16 * K_ * R + 16 * K_ * 2 + 16 * 194) * sizeof(float);   \
        cnf_span_kernel<S><<<6 * (SEQ - S + 1), 192, shmem, stream>>>(        \
            rulesP, chart);                                                   \
        break;                                                                \
    }

    for (int span = 2; span <= SEQ; ++span) {
        switch (span) {
            LAUNCH_SPAN(2)  LAUNCH_SPAN(3)  LAUNCH_SPAN(4)  LAUNCH_SPAN(5)
            LAUNCH_SPAN(6)  LAUNCH_SPAN(7)  LAUNCH_SPAN(8)  LAUNCH_SPAN(9)
            LAUNCH_SPAN(10) LAUNCH_SPAN(11) LAUNCH_SPAN(12) LAUNCH_SPAN(13)
            LAUNCH_SPAN(14) LAUNCH_SPAN(15) LAUNCH_SPAN(16) LAUNCH_SPAN(17)
            LAUNCH_SPAN(18) LAUNCH_SPAN(19) LAUNCH_SPAN(20) LAUNCH_SPAN(21)
            LAUNCH_SPAN(22) LAUNCH_SPAN(23) LAUNCH_SPAN(24)
        }
    }
#undef LAUNCH_SPAN

    cnf_finalize<<<1, R, 0, stream>>>(start_logits, chart, out);
}